// AGATLayer_43190191128612
// MI455X (gfx1250) — compile-verified
//
#include <hip/hip_runtime.h>
#include <math.h>

#define N_NODES 100000
#define N_EDGES 3200000
#define IN_DIM  62
#define OUT_DIM 16

typedef float v2f __attribute__((ext_vector_type(2)));
typedef float v8f __attribute__((ext_vector_type(8)));

// ---------------------------------------------------------------------------
// Kernel 1: z = h @ W_fc^T  via V_WMMA_F32_16X16X4_F32
// One wave (32 lanes) computes one 16x16 tile: 16 rows of h x all 16 outputs.
// K = 62 padded to 64 -> 16 WMMA steps of K=4.
// A (16x4 f32): lanes 0-15 hold K=k0,k0+1 in v[0],v[1]; lanes 16-31 hold K=k0+2,k0+3.
// B (4x16 f32): lane n<16 holds B[k0][n],B[k0+1][n]; lanes 16-31 hold B[k0+2..3][n-16].
// B[k][n] = W_fc[n][k] (row-major W_fc is OUT_DIM x IN_DIM).
// ---------------------------------------------------------------------------
__global__ void agat_gemm_z(const float* __restrict__ h,
                            const float* __restrict__ Wfc,
                            float* __restrict__ z) {
  const int wave = (blockIdx.x * blockDim.x + threadIdx.x) >> 5;
  const int lane = threadIdx.x & 31;
  const int n_tiles = N_NODES / 16;            // 6250 exactly
  if (wave >= n_tiles) return;                 // wave-uniform: EXEC stays all-1s

  const int m    = lane & 15;                  // row-in-tile (A) / column (B)
  const bool hi  = lane >= 16;
  const int row  = wave * 16 + m;              // h row for A fragment

  v8f acc = {0.f, 0.f, 0.f, 0.f, 0.f, 0.f, 0.f, 0.f};

  const float* hrow = h + (long)row * IN_DIM;
  const float* wrow = Wfc + (long)m * IN_DIM;  // W_fc[n=m][*]

#pragma unroll
  for (int k0 = 0; k0 < 64; k0 += 4) {
    const int ka = k0 + (hi ? 2 : 0);
    v2f a, b;
    a.x = (ka     < IN_DIM) ? hrow[ka]     : 0.f;
    a.y = (ka + 1 < IN_DIM) ? hrow[ka + 1] : 0.f;
    b.x = (ka     < IN_DIM) ? wrow[ka]     : 0.f;
    b.y = (ka + 1 < IN_DIM) ? wrow[ka + 1] : 0.f;
    acc = __builtin_amdgcn_wmma_f32_16x16x4_f32(
        /*neg_a=*/false, a, /*neg_b=*/false, b,
        /*c_mod=*/(short)0, acc, /*reuse_a=*/false, /*reuse_b=*/false);
  }

  // D layout: VGPR r -> row (r [+8 for hi lanes]) of tile, column = m
  const int rbase = wave * 16 + (hi ? 8 : 0);
#pragma unroll
  for (int r = 0; r < 8; ++r)
    z[(long)(rbase + r) * OUT_DIM + m] = acc[r];
}

// ---------------------------------------------------------------------------
// Kernel 2: per-node attention scalars a_src = z.Wa_src, a_dst = z.Wa_dst,
// and init of segment-max / denom / output buffers.
// ---------------------------------------------------------------------------
__global__ void agat_node(const float* __restrict__ z,
                          const float* __restrict__ Wattn,   // 34 floats
                          float* __restrict__ a_src,
                          float* __restrict__ a_dst,
                          float* __restrict__ mmax,
                          float* __restrict__ denom,
                          float* __restrict__ out) {
  const int n = blockIdx.x * blockDim.x + threadIdx.x;
  if (n >= N_NODES) return;
  float s = 0.f, d = 0.f;
  const float* zr = z + (long)n * OUT_DIM;
#pragma unroll
  for (int j = 0; j < OUT_DIM; ++j) {
    const float zj = zr[j];
    s += zj * Wattn[j];
    d += zj * Wattn[OUT_DIM + j];
  }
  a_src[n] = s;
  a_dst[n] = d;
  mmax[n]  = -INFINITY;
  denom[n] = 0.f;
#pragma unroll
  for (int j = 0; j < OUT_DIM; ++j) out[(long)n * OUT_DIM + j] = 0.f;
}

// Monotone float atomic-max via sign-split int atomics (always available).
__device__ inline void atomicMaxF32(float* addr, float val) {
  const int vi = __float_as_int(val);
  if (vi >= 0) atomicMax((int*)addr, vi);
  else         atomicMin((unsigned int*)addr, (unsigned int)vi);
}

// ---------------------------------------------------------------------------
// Kernel 3: per-edge raw score + leaky-relu, atomic segment-max over dst.
// ex = e @ W_edge^T recomputed on the fly (4 FLOPs, cheaper than 25.6MB ws).
// ---------------------------------------------------------------------------
__global__ void agat_edge_att(const float* __restrict__ e,
                              const int* __restrict__ src,
                              const int* __restrict__ dst,
                              const float* __restrict__ a_src,
                              const float* __restrict__ a_dst,
                              const float* __restrict__ Wattn,
                              const float* __restrict__ Wedge,  // 2x2 row-major
                              float* __restrict__ eatt,
                              float* __restrict__ mmax) {
  const int i = blockIdx.x * blockDim.x + threadIdx.x;
  if (i >= N_EDGES) return;
  const float e0 = e[2 * i], e1 = e[2 * i + 1];
  const float ex0 = e0 * Wedge[0] + e1 * Wedge[1];
  const float ex1 = e0 * Wedge[2] + e1 * Wedge[3];
  const int di = dst[i];
  float a = a_src[src[i]] + a_dst[di] + ex0 * Wattn[32] + ex1 * Wattn[33];
  a = (a > 0.f) ? a : 0.01f * a;               // leaky relu
  eatt[i] = a;
  atomicMaxF32(&mmax[di], a);
}

// ---------------------------------------------------------------------------
// Kernel 4: w = exp(att - m[dst]); denom[dst] += w  (w overwrites eatt buffer)
// ---------------------------------------------------------------------------
__global__ void agat_edge_w(const int* __restrict__ dst,
                            const float* __restrict__ mmax,
                            float* __restrict__ w,
                            float* __restrict__ denom) {
  const int i = blockIdx.x * blockDim.x + threadIdx.x;
  if (i >= N_EDGES) return;
  const int di = dst[i];
  const float wi = __expf(w[i] - mmax[di]);
  w[i] = wi;
  atomicAdd(&denom[di], wi);
}

// ---------------------------------------------------------------------------
// Kernel 5: 16 lanes per edge: out[dst] += (w/denom[dst]) * (z[src] + ez)
// ez = ex @ W_e2n^T  (per-output-channel 2-FLOP). Coalesced z reads+atomics.
// ---------------------------------------------------------------------------
__global__ void agat_edge_msg(const float* __restrict__ e,
                              const int* __restrict__ src,
                              const int* __restrict__ dst,
                              const float* __restrict__ z,
                              const float* __restrict__ w,
                              const float* __restrict__ denom,
                              const float* __restrict__ Wedge,
                              const float* __restrict__ We2n,  // 16x2 row-major
                              float* __restrict__ out) {
  const int tid = blockIdx.x * blockDim.x + threadIdx.x;
  const int i = tid >> 4;          // edge
  const int j = tid & 15;          // output channel
  if (i >= N_EDGES) return;
  const int s  = src[i];
  const int di = dst[i];
  const float attn = w[i] / denom[di];
  const float e0 = e[2 * i], e1 = e[2 * i + 1];
  const float ex0 = e0 * Wedge[0] + e1 * Wedge[1];
  const float ex1 = e0 * Wedge[2] + e1 * Wedge[3];
  const float ez  = ex0 * We2n[2 * j] + ex1 * We2n[2 * j + 1];
  const float msg = attn * (z[(long)s * OUT_DIM + j] + ez);
  atomicAdd(&out[(long)di * OUT_DIM + j], msg);
}

// ---------------------------------------------------------------------------
extern "C" void kernel_launch(void* const* d_in, const int* in_sizes, int n_in,
                              void* d_out, int out_size, void* d_ws, size_t ws_size,
                              hipStream_t stream) {
  const float* h     = (const float*)d_in[0];
  const float* e     = (const float*)d_in[1];
  const int*   src   = (const int*)  d_in[2];
  const int*   dst   = (const int*)  d_in[3];
  const float* Wfc   = (const float*)d_in[4];
  const float* Wattn = (const float*)d_in[5];
  const float* Wedge = (const float*)d_in[6];
  const float* We2n  = (const float*)d_in[7];
  float* out = (float*)d_out;

  // workspace carve-up (~20.8 MB)
  char* ws = (char*)d_ws;
  float* z      = (float*)ws;                       ws += (size_t)N_NODES * OUT_DIM * sizeof(float);
  float* a_src  = (float*)ws;                       ws += (size_t)N_NODES * sizeof(float);
  float* a_dst  = (float*)ws;                       ws += (size_t)N_NODES * sizeof(float);
  float* mmax   = (float*)ws;                       ws += (size_t)N_NODES * sizeof(float);
  float* denom  = (float*)ws;                       ws += (size_t)N_NODES * sizeof(float);
  float* w      = (float*)ws;                       ws += (size_t)N_EDGES * sizeof(float);
  (void)in_sizes; (void)n_in; (void)out_size; (void)ws_size;

  const int B = 256;

  // 1) z = h @ Wfc^T (WMMA): 6250 waves, 8 waves per block
  {
    const int n_waves  = N_NODES / 16;
    const int blocks   = (n_waves * 32 + B - 1) / B;
    agat_gemm_z<<<blocks, B, 0, stream>>>(h, Wfc, z);
  }
  // 2) node scalars + init
  agat_node<<<(N_NODES + B - 1) / B, B, 0, stream>>>(z, Wattn, a_src, a_dst, mmax, denom, out);
  // 3) edge scores + segment max
  agat_edge_att<<<(N_EDGES + B - 1) / B, B, 0, stream>>>(e, src, dst, a_src, a_dst, Wattn, Wedge, w, mmax);
  // 4) exp + segment sum
  agat_edge_w<<<(N_EDGES + B - 1) / B, B, 0, stream>>>(dst, mmax, w, denom);
  // 5) weighted scatter-add, 16 lanes per edge
  {
    const long threads = (long)N_EDGES * 16;
    const int blocks = (int)((threads + B - 1) / B);
    agat_edge_msg<<<blocks, B, 0, stream>>>(e, src, dst, z, w, denom, Wedge, We2n, out);
  }
}